// VSALattice_30726196035983
// MI455X (gfx1250) — compile-verified
//
#include <hip/hip_runtime.h>
#include <math.h>

// Problem constants (from reference)
#define Bsz   64
#define Ssz   128
#define Rsz   8
#define Dsz   10000
#define Asz   10
#define Psz   256
#define Fbins 5001            // D/2 + 1
#define NC    10016           // 2*Fbins (=10002) padded to multiple of 16
#define XROWS 144             // 139 (128 pos + 10 atom + 1 closure) padded to 16
#define TWO_PI 6.2831853071795864769f
#define PHSC  (TWO_PI / (float)Dsz)   // compile-time constant: no runtime divide

#define TILE_K   256                  // TDM chunk width (elements)
#define LDS_PITCH 257                 // 256 data + 1 pad dword (bank-conflict-free)

typedef float v2f __attribute__((ext_vector_type(2)));
typedef float v8f __attribute__((ext_vector_type(8)));
typedef unsigned int ui32x4 __attribute__((ext_vector_type(4)));
typedef int i32x4 __attribute__((ext_vector_type(4)));
typedef int i32x8 __attribute__((ext_vector_type(8)));

// ---------------------------------------------------------------------------
// Tensor Data Mover: 2D tile load Global -> LDS.
//   tile_w x tile_h elements (f32), row stride row_stride_elems.
//   remaining_w/remaining_h bound the tensor from the tile origin: OOB reads
//   return zero (zero-fills the tail chunk -> safe GEMM padding).
//   pad_enable: +1 dword after every 256 dwords => LDS row pitch 257.
// ---------------------------------------------------------------------------
__device__ __forceinline__ void tdm_load_tile_2d(unsigned lds_off,
                                                 const float* gsrc,
                                                 unsigned remaining_w,
                                                 unsigned remaining_h,
                                                 unsigned tile_w,
                                                 unsigned tile_h,
                                                 unsigned row_stride_elems) {
  unsigned long long ga = (unsigned long long)gsrc;
  ui32x4 g0;
  g0.x = 1u;                                            // count=1 (user D#)
  g0.y = lds_off;                                       // lds_addr (bytes)
  g0.z = (unsigned)ga;                                  // global_addr[31:0]
  g0.w = (unsigned)((ga >> 32) & 0x01FFFFFFull)         // global_addr[56:32]
         | 0x80000000u;                                 // type=2 ("image")
  i32x8 g1;
  g1[0] = (int)(0x00020000u                             // data_size=4B
                | (1u << 20)                            // pad_enable
                | (7u << 22));                          // pad_interval=256 DW
                                                        // pad_amount=0 -> 1 DW
  g1[1] = (int)((remaining_w & 0xFFFFu) << 16);         // tensor_dim0[15:0]
  g1[2] = (int)(((remaining_w >> 16) & 0xFFFFu)         // tensor_dim0[31:16]
                | ((remaining_h & 0xFFFFu) << 16));     // tensor_dim1[15:0]
  g1[3] = (int)(((remaining_h >> 16) & 0xFFFFu)         // tensor_dim1[31:16]
                | ((tile_w & 0xFFFFu) << 16));          // tile_dim0
  g1[4] = (int)(tile_h & 0xFFFFu);                      // tile_dim1 (tile_dim2=0)
  g1[5] = (int)row_stride_elems;                        // tensor_dim0_stride lo
  g1[6] = 0;
  g1[7] = 0;
  i32x4 gz = {};                                        // 2D: groups 2/3 unused
#if __clang_major__ >= 23
  i32x8 gz8 = {};
  __builtin_amdgcn_tensor_load_to_lds(g0, g1, gz, gz, gz8, 0);
#else
  __builtin_amdgcn_tensor_load_to_lds(g0, g1, gz, gz, 0);
#endif
}

// ---------------------------------------------------------------------------
// K0: pack pos/atom/closure rows into one padded matrix X[XROWS][Dsz]
// ---------------------------------------------------------------------------
__global__ void pack_rows_kernel(const float* __restrict__ atom,
                                 const float* __restrict__ pos,
                                 const float* __restrict__ clos,
                                 float* __restrict__ X) {
  int i = blockIdx.x * blockDim.x + threadIdx.x;
  if (i >= XROWS * Dsz) return;
  int r = i / Dsz, d = i - r * Dsz;
  float v = 0.0f;
  if (r < Ssz)                 v = pos[r * Dsz + d];
  else if (r < Ssz + Asz)      v = atom[(r - Ssz) * Dsz + d];
  else if (r == Ssz + Asz)     v = clos[d];
  X[i] = v;
}

// ---------------------------------------------------------------------------
// K1: forward DFT as GEMM.  FT[m, n] = sum_d X[m,d] * basis(n, d)
//     basis(n, d) = (n odd ? -sin : cos)(2*pi*(n>>1)*d / D)
//     Block = 4 waves sharing one M-tile (blockIdx.y); each wave owns one
//     N-tile.  A tile staged by TDM into LDS (pitch 257); per-lane basis by
//     phasor recurrence with exact integer-phase resync every 128 steps.
// ---------------------------------------------------------------------------
__global__ void dft_fwd_kernel(const float* __restrict__ X,
                               float* __restrict__ FT) {
  __shared__ float tileA[16 * LDS_PITCH];
  const int lane = threadIdx.x & 31;
  const int wv = threadIdx.x >> 5;
  const int NT = NC / 16;                  // 626
  const int mt = blockIdx.y;
  const int nt = blockIdx.x * 4 + wv;      // may exceed NT-1: store is guarded
  const int m0 = mt * 16, n0 = nt * 16;

  const int half = (lane < 16) ? 0 : 2;    // K offset of this lane half
  const int ncol = n0 + (lane & 15);       // B-matrix column (N)
  const int kbin = ncol >> 1;
  const bool isSin = (ncol & 1) != 0;

  float c1, s1, c4, s4;
  __sincosf((float)(kbin % Dsz) * PHSC, &s1, &c1);          // +theta
  __sincosf((float)((4 * kbin) % Dsz) * PHSC, &s4, &c4);    // +4*theta
  float cp, sp;                                             // phasor at d
  __sincosf((float)((kbin * half) % Dsz) * PHSC, &sp, &cp);

  v8f acc = {};
  const int aoff = (lane & 15) * LDS_PITCH + half;
  int step = 0;
  for (int kk = 0; kk < Dsz; kk += TILE_K) {
    __syncthreads();                       // previous chunk fully consumed
    if (threadIdx.x < 32) {
      tdm_load_tile_2d((unsigned)(unsigned long long)(void*)tileA,
                       X + (size_t)m0 * Dsz + kk,
                       (unsigned)(Dsz - kk), (unsigned)(XROWS - m0),
                       TILE_K, 16u, (unsigned)Dsz);
      __builtin_amdgcn_s_wait_tensorcnt(0);
    }
    __syncthreads();                       // tile visible to all waves
    const int kend = (kk + TILE_K < Dsz) ? (kk + TILE_K) : Dsz;
    for (int k2 = kk; k2 < kend; k2 += 4, ++step) {
      if ((step & 127) == 0) {             // exact phase resync
        int m = (kbin * (k2 + half)) % Dsz;
        __sincosf((float)m * PHSC, &sp, &cp);
      }
      v2f b;
      b.x = isSin ? -sp : cp;                      // basis at d = k2+half
      float c2 = cp * c1 - sp * s1;                // rotate +theta -> d+1
      float s2 = sp * c1 + cp * s1;
      b.y = isSin ? -s2 : c2;

      v2f a;
      a.x = tileA[aoff + (k2 - kk)];
      a.y = tileA[aoff + (k2 - kk) + 1];

      acc = __builtin_amdgcn_wmma_f32_16x16x4_f32(false, a, false, b,
                                                  (short)0, acc, false, false);
      float cn = cp * c4 - sp * s4;                // advance phasor by 4*theta
      sp = sp * c4 + cp * s4;
      cp = cn;
    }
  }
  if (nt < NT) {
#pragma unroll
    for (int r = 0; r < 8; ++r) {
      int row = m0 + r + ((lane < 16) ? 0 : 8);
      FT[row * NC + n0 + (lane & 15)] = acc[r];
    }
  }
}

// ---------------------------------------------------------------------------
// K2: frequency-domain gather/bind/bundle.  One thread per (b, k).
// ---------------------------------------------------------------------------
__global__ void assemble_kernel(const float* __restrict__ FT,
                                const float* __restrict__ token_mask,
                                const float* __restrict__ ring_mask,
                                const int* __restrict__ atom_idx,
                                const int* __restrict__ ring_pos,
                                float* __restrict__ FREQ) {
  const int KP = NC / 2;  // 5008
  int tid = blockIdx.x * blockDim.x + threadIdx.x;
  if (tid >= Bsz * KP) return;
  int b = tid / KP, k = tid - b * KP;
  float* out = FREQ + b * NC + 2 * k;
  if (k > Fbins - 1) { out[0] = 0.0f; out[1] = 0.0f; return; }

  const int col = 2 * k;
  float re = 0.0f, im = 0.0f;
  for (int s = 0; s < Ssz; ++s) {
    float m = token_mask[b * Ssz + s];
    int a = atom_idx[b * Ssz + s];
    const float* fa = FT + (Ssz + a) * NC + col;
    const float* fp = FT + s * NC + col;
    float ar = fa[0], ai = fa[1], pr = fp[0], pi = fp[1];
    re += m * (ar * pr - ai * pi);
    im += m * (ar * pi + ai * pr);
  }
  const float* fc = FT + (Ssz + Asz) * NC + col;
  float cr = fc[0], ci = fc[1];
  for (int r = 0; r < Rsz; ++r) {
    float rm = ring_mask[b * Rsz + r];
    int i0 = ring_pos[(b * Rsz + r) * 2 + 0];
    int i1 = ring_pos[(b * Rsz + r) * 2 + 1];
    const float* p0 = FT + i0 * NC + col;
    const float* p1 = FT + i1 * NC + col;
    float xr = p0[0] * p1[0] - p0[1] * p1[1];
    float xi = p0[0] * p1[1] + p0[1] * p1[0];
    re += rm * (xr * cr - xi * ci);
    im += rm * (xr * ci + xi * cr);
  }
  float w = (k == 0 || k == Fbins - 1) ? (1.0f / (float)Dsz) : (2.0f / (float)Dsz);
  out[0] = w * re;
  out[1] = w * im;
}

// ---------------------------------------------------------------------------
// K3: inverse DFT as GEMM.  MOL[b, d] = sum_n FREQ[b,n] * basis(n, d)
//     Same TDM-staged A pattern; lane's phasor advances by 2*pi*2d/D per step.
// ---------------------------------------------------------------------------
__global__ void dft_inv_kernel(const float* __restrict__ FREQ,
                               float* __restrict__ MOL) {
  __shared__ float tileA[16 * LDS_PITCH];
  const int lane = threadIdx.x & 31;
  const int wv = threadIdx.x >> 5;
  const int NT = Dsz / 16;                 // 625
  const int mt = blockIdx.y;
  const int nt = blockIdx.x * 4 + wv;      // may exceed NT-1: store is guarded
  const int m0 = mt * 16, n0 = nt * 16;

  const int half = (lane < 16) ? 0 : 2;
  const int k0 = half >> 1;                // 0 or 1: lane's kbin offset
  const int d = n0 + (lane & 15);

  float c2r, s2r;                          // rotate by delta_kbin = 2
  __sincosf((float)((2 * d) % Dsz) * PHSC, &s2r, &c2r);
  float cp, sp;
  __sincosf((float)((k0 * d) % Dsz) * PHSC, &sp, &cp);

  v8f acc = {};
  const int aoff = (lane & 15) * LDS_PITCH + half;
  int step = 0;
  for (int kk = 0; kk < NC; kk += TILE_K) {
    __syncthreads();
    if (threadIdx.x < 32) {
      tdm_load_tile_2d((unsigned)(unsigned long long)(void*)tileA,
                       FREQ + (size_t)m0 * NC + kk,
                       (unsigned)(NC - kk), (unsigned)(Bsz - m0),
                       TILE_K, 16u, (unsigned)NC);
      __builtin_amdgcn_s_wait_tensorcnt(0);
    }
    __syncthreads();
    const int kend = (kk + TILE_K < NC) ? (kk + TILE_K) : NC;
    for (int k2 = kk; k2 < kend; k2 += 4, ++step) {
      if ((step & 127) == 0) {
        int kb = (k2 >> 1) + k0;
        int m = (kb * d) % Dsz;
        __sincosf((float)m * PHSC, &sp, &cp);
      }
      v2f b;
      b.x = cp;        // even column: cos
      b.y = -sp;       // odd column : -sin (same kbin)
      v2f a;
      a.x = tileA[aoff + (k2 - kk)];
      a.y = tileA[aoff + (k2 - kk) + 1];

      acc = __builtin_amdgcn_wmma_f32_16x16x4_f32(false, a, false, b,
                                                  (short)0, acc, false, false);
      float cn = cp * c2r - sp * s2r;
      sp = sp * c2r + cp * s2r;
      cp = cn;
    }
  }
  if (nt < NT) {
#pragma unroll
    for (int r = 0; r < 8; ++r) {
      int row = m0 + r + ((lane < 16) ? 0 : 8);
      MOL[row * Dsz + n0 + (lane & 15)] = acc[r];
    }
  }
}

// ---------------------------------------------------------------------------
// K4: projection GEMM.  OUT[b, p] = sum_d MOL[b,d] * W[p,d] + bias[p]
//     One block per 16-wide p tile; wave w owns M-tile w.  W tile (B operand)
//     staged by TDM; A streams from global with prefetch.
// ---------------------------------------------------------------------------
__global__ void proj_kernel(const float* __restrict__ MOL,
                            const float* __restrict__ W,
                            const float* __restrict__ bias,
                            float* __restrict__ OUT) {
  __shared__ float tileB[16 * LDS_PITCH];
  const int lane = threadIdx.x & 31;
  const int wv = threadIdx.x >> 5;
  const int m0 = wv * 16;                  // 4 waves cover all of B=64
  const int n0 = blockIdx.x * 16;          // p tile

  const int half = (lane < 16) ? 0 : 2;
  const int mrow = m0 + (lane & 15);
  const int p = n0 + (lane & 15);

  v8f acc = {};
  const int boff = (lane & 15) * LDS_PITCH + half;
  for (int kk = 0; kk < Dsz; kk += TILE_K) {
    __syncthreads();
    if (threadIdx.x < 32) {
      tdm_load_tile_2d((unsigned)(unsigned long long)(void*)tileB,
                       W + (size_t)n0 * Dsz + kk,
                       (unsigned)(Dsz - kk), (unsigned)(Psz - n0),
                       TILE_K, 16u, (unsigned)Dsz);
      __builtin_amdgcn_s_wait_tensorcnt(0);
    }
    __syncthreads();
    const int kend = (kk + TILE_K < Dsz) ? (kk + TILE_K) : Dsz;
    const float* ap = MOL + (size_t)mrow * Dsz + kk + half;
    int step = 0;
    for (int k2 = kk; k2 < kend; k2 += 4, ++step) {
      if ((step & 31) == 0) __builtin_prefetch(ap + 256, 0, 0);
      v2f a, b;
      a.x = ap[0]; a.y = ap[1]; ap += 4;
      b.x = tileB[boff + (k2 - kk)];
      b.y = tileB[boff + (k2 - kk) + 1];
      acc = __builtin_amdgcn_wmma_f32_16x16x4_f32(false, a, false, b,
                                                  (short)0, acc, false, false);
    }
  }
  float bb = bias[p];
#pragma unroll
  for (int r = 0; r < 8; ++r) {
    int row = m0 + r + ((lane < 16) ? 0 : 8);
    OUT[row * Psz + n0 + (lane & 15)] = acc[r] + bb;
  }
}

// ---------------------------------------------------------------------------
// Launcher
// ---------------------------------------------------------------------------
extern "C" void kernel_launch(void* const* d_in, const int* in_sizes, int n_in,
                              void* d_out, int out_size, void* d_ws, size_t ws_size,
                              hipStream_t stream) {
  const float* atom  = (const float*)d_in[0];   // [A, D]
  const float* pos   = (const float*)d_in[1];   // [S, D]
  const float* clos  = (const float*)d_in[2];   // [D]
  const float* W     = (const float*)d_in[3];   // [P, D]
  const float* bias  = (const float*)d_in[4];   // [P]
  const float* tmask = (const float*)d_in[5];   // [B, S]
  const float* rmask = (const float*)d_in[6];   // [B, R]
  const int*   aidx  = (const int*)d_in[7];     // [B, S]
  const int*   rpos  = (const int*)d_in[8];     // [B, R, 2]
  float* OUT = (float*)d_out;                   // [B, P]

  float* ws   = (float*)d_ws;
  float* X    = ws;                       // XROWS * Dsz
  float* FT   = X + (size_t)XROWS * Dsz;  // XROWS * NC
  float* FREQ = FT + (size_t)XROWS * NC;  // Bsz * NC
  float* MOL  = FREQ + (size_t)Bsz * NC;  // Bsz * Dsz

  {
    int n = XROWS * Dsz;
    pack_rows_kernel<<<(n + 255) / 256, 256, 0, stream>>>(atom, pos, clos, X);
  }
  {
    dim3 grid((NC / 16 + 3) / 4, XROWS / 16);   // (157, 9)
    dft_fwd_kernel<<<grid, 128, 0, stream>>>(X, FT);
  }
  {
    int n = Bsz * (NC / 2);
    assemble_kernel<<<(n + 255) / 256, 256, 0, stream>>>(FT, tmask, rmask, aidx, rpos, FREQ);
  }
  {
    dim3 grid((Dsz / 16 + 3) / 4, Bsz / 16);    // (157, 4)
    dft_inv_kernel<<<grid, 128, 0, stream>>>(FREQ, MOL);
  }
  {
    proj_kernel<<<Psz / 16, 128, 0, stream>>>(MOL, W, bias, OUT);
  }
}